// SNNNet_28621662060586
// MI455X (gfx1250) — compile-verified
//
#include <hip/hip_runtime.h>

typedef __attribute__((ext_vector_type(2))) float v2f;
typedef __attribute__((ext_vector_type(8))) float v8f;

#define B_SZ    4096
#define N_INP   1024
#define N_HID   256
#define T_STEPS 1024   // reference loops over x.shape[1]
#define BETA    0.9f
#define THR     1.0f

// ---------------------------------------------------------------------------
// Phase 1: cur1 = x @ W1^T + b1  via V_WMMA_F32_16X16X4_F32 (exact f32 math).
// grid.x = B/16 M-tiles; block = 128 threads = 4 waves; wave w covers columns
// [w*64, w*64+64) (4 N-tiles of 16), reusing the A fragment across 4 WMMAs.
//
// f32 16x4 A layout: lanes 0-15 -> M=0..15, K={k,k+1}; lanes 16-31 -> K={k+2,k+3}
// f32 4x16 B layout (mirrored): lane&15 -> N, lane>>4 selects K pair.
// C/D 16x16 f32: VGPR r, lane l -> M = r + 8*(l>>4), N = l&15.
// ---------------------------------------------------------------------------
__global__ __launch_bounds__(128) void snn_gemm_wmma(
    const float* __restrict__ x, const float* __restrict__ W1,
    const float* __restrict__ b1, float* __restrict__ cur1)
{
    const int lane = threadIdx.x & 31;
    const int wid  = threadIdx.x >> 5;
    const int m    = lane & 15;
    const int half = lane >> 4;                 // 0/1 selects K pair
    const int mrow = blockIdx.x * 16 + m;

    const float* Arow = x + (size_t)mrow * N_INP + 2 * half;

    int ncol0 = wid * 64 + m;
    const float* B0 = W1 + (size_t)(ncol0 +  0) * N_INP + 2 * half;
    const float* B1 = W1 + (size_t)(ncol0 + 16) * N_INP + 2 * half;
    const float* B2 = W1 + (size_t)(ncol0 + 32) * N_INP + 2 * half;
    const float* B3 = W1 + (size_t)(ncol0 + 48) * N_INP + 2 * half;

    v8f acc0 = {}, acc1 = {}, acc2 = {}, acc3 = {};

    for (int k = 0; k < N_INP; k += 4) {
        v2f a   = *(const v2f*)(Arow + k);
        v2f b0  = *(const v2f*)(B0 + k);
        v2f b1v = *(const v2f*)(B1 + k);
        v2f b2v = *(const v2f*)(B2 + k);
        v2f b3v = *(const v2f*)(B3 + k);
        acc0 = __builtin_amdgcn_wmma_f32_16x16x4_f32(false, a, false, b0,  (short)0, acc0, false, false);
        acc1 = __builtin_amdgcn_wmma_f32_16x16x4_f32(false, a, false, b1v, (short)0, acc1, false, false);
        acc2 = __builtin_amdgcn_wmma_f32_16x16x4_f32(false, a, false, b2v, (short)0, acc2, false, false);
        acc3 = __builtin_amdgcn_wmma_f32_16x16x4_f32(false, a, false, b3v, (short)0, acc3, false, false);
    }

    // Add bias, store D tiles. Column per tile = ncol0 + 16*nt; row = base + r + 8*half.
    const int rbase = blockIdx.x * 16 + 8 * half;
    const float bb0 = b1[ncol0 +  0];
    const float bb1 = b1[ncol0 + 16];
    const float bb2 = b1[ncol0 + 32];
    const float bb3 = b1[ncol0 + 48];
    #pragma unroll
    for (int r = 0; r < 8; ++r) {
        float* orow = cur1 + (size_t)(rbase + r) * N_HID + ncol0;
        orow[ 0] = acc0[r] + bb0;
        orow[16] = acc1[r] + bb1;
        orow[32] = acc2[r] + bb2;
        orow[48] = acc3[r] + bb3;
    }
}

// ---------------------------------------------------------------------------
// Phase 2: sequential T-loop. One wave per batch row; 8 hid elements per lane
// held in registers. Key identity exploited: the spike compare on mem1_new at
// step t IS the reset compare on mem1_old at step t+1, so the comparison
// result `sel` is carried across iterations instead of recomputed (saves 8
// v_cmp per step in the dominant loop). Masked-W2 dot reduced with 5
// xor-shuffles; mem2 recurrence replicated across lanes; lane 0 stores.
// ---------------------------------------------------------------------------
__global__ __launch_bounds__(256) void snn_time_loop(
    const float* __restrict__ cur1, const float* __restrict__ W2,
    const float* __restrict__ b2, float* __restrict__ out)
{
    const int lane = threadIdx.x & 31;
    const int wv   = threadIdx.x >> 5;
    const int b    = blockIdx.x * 8 + wv;     // batch row
    const int hb   = lane * 8;                // first hid index for this lane

    const float* c = cur1 + (size_t)b * N_HID + hb;
    float cu[8], cm[8], w[8], mem1[8];
    bool  sel[8];
    #pragma unroll
    for (int j = 0; j < 8; ++j) {
        cu[j]   = c[j];
        cm[j]   = cu[j] - THR;                // cur1 - reset*THR (reset==1 case)
        w[j]    = W2[hb + j];
        mem1[j] = 0.0f;
        sel[j]  = false;                      // heaviside(0 - THR) == 0
    }
    const float b2v = b2[0];
    float mem2 = 0.0f;
    bool  sel2 = false;

    float* __restrict__ outm = out;                               // mem2_rec [T,B]
    float* __restrict__ outs = out + (size_t)T_STEPS * B_SZ;      // spk2_rec [T,B]

    for (int t = 0; t < T_STEPS; ++t) {
        float partial = 0.0f;
        #pragma unroll
        for (int j = 0; j < 8; ++j) {
            float inc = sel[j] ? cm[j] : cu[j];   // subtract reset from prev step
            mem1[j]   = fmaf(BETA, mem1[j], inc);
            sel[j]    = mem1[j] > THR;            // spike now, reset next step
            partial  += sel[j] ? w[j] : 0.0f;
        }
        #pragma unroll
        for (int off = 16; off > 0; off >>= 1)
            partial += __shfl_xor(partial, off, 32);

        float cur2 = partial + b2v;
        float r2   = sel2 ? THR : 0.0f;
        mem2       = fmaf(BETA, mem2, cur2) - r2;
        sel2       = mem2 > THR;
        float spk2 = sel2 ? 1.0f : 0.0f;

        if (lane == 0) {
            outm[(size_t)t * B_SZ + b] = mem2;
            outs[(size_t)t * B_SZ + b] = spk2;
        }
    }
}

// ---------------------------------------------------------------------------
extern "C" void kernel_launch(void* const* d_in, const int* in_sizes, int n_in,
                              void* d_out, int out_size, void* d_ws, size_t ws_size,
                              hipStream_t stream) {
    const float* x  = (const float*)d_in[0];
    const float* W1 = (const float*)d_in[1];
    const float* b1 = (const float*)d_in[2];
    const float* W2 = (const float*)d_in[3];
    const float* b2 = (const float*)d_in[4];
    float* out  = (float*)d_out;
    float* cur1 = (float*)d_ws;   // 4096*256 f32 = 4 MB scratch

    snn_gemm_wmma<<<dim3(B_SZ / 16), dim3(128), 0, stream>>>(x, W1, b1, cur1);
    snn_time_loop<<<dim3(B_SZ / 8), dim3(256), 0, stream>>>(cur1, W2, b2, out);
}